// MFCC_49993419325512
// MI455X (gfx1250) — compile-verified
//
#include <hip/hip_runtime.h>
#include <math.h>

typedef __attribute__((ext_vector_type(2))) float v2f;
typedef __attribute__((ext_vector_type(8))) float v8f;

#define SRATE   16000
#define FLEN    400
#define FSTEP   160
#define NFFT    512
#define NBINS   257      // NFFT/2 + 1
#define NFILT   26
#define NUMCEP  13
#define BATCH   64
#define TLEN    320000
#define NFRAMES 1999     // 1 + ceil((320000-400)/160)
#define FTILES  125      // ceil(1999/16)
#define NTILE   16       // freq tiles of 16 -> bins 0..255 (bin 256 = Nyquist, VALU)
#define KSTEPS  100      // 400 / 4 (K per f32 WMMA)
#define FBSTEPS 65       // 260 / 4 (pspec K padded 257->260)
#define DCSTEPS 7        // 28 / 4  (feat K padded 26->28)
#define EPSF    1.1920929e-07f
#define PREEMPH 0.97f
#define FROW    404      // LDS frame row stride (floats), conflict-free vs 64 banks
#define PROW    260      // LDS pspec row stride (= padded K for fb GEMM)

__device__ __forceinline__ double mel_bin(int m) {
  double melmax = 2595.0 * log10(1.0 + 8000.0 / 700.0);
  double mel    = melmax * (double)m / (double)(NFILT + 1);
  double hz     = 700.0 * (pow(10.0, mel / 2595.0) - 1.0);
  return floor((double)(NFFT + 1) * hz / (double)SRATE);
}

__device__ __forceinline__ float fb_val(int i, int j) {  // [bin i][filter j]
  double b0 = mel_bin(j), b1 = mel_bin(j + 1), b2 = mel_bin(j + 2);
  double di = (double)i;
  if (di >= b0 && di < b1) return (float)((di - b0) / (b1 - b0));
  if (di >= b1 && di < b2) return (float)((b2 - di) / (b2 - b1));
  return 0.f;
}

__device__ __forceinline__ float dct_val(int k, int n) {  // [cep k][filter n]
  const double PI = 3.14159265358979323846;
  double c  = cos(PI * (double)(2 * n + 1) * (double)k / (2.0 * NFILT));
  double sc = (k == 0) ? (1.0 / sqrt((double)NFILT)) : sqrt(2.0 / (double)NFILT);
  return (float)(c * sc);
}

// All constant matrices are emitted pre-swizzled into WMMA B-fragment order:
//   element e = ((tile*steps + step)*32 + lane)*2 + j
//   lane -> N = tile*16 + (lane&15), K = step*4 + (lane>=16 ? 2 : 0) + j
__global__ void mfcc_setup(float* __restrict__ cosw, float* __restrict__ sinw,
                           float* __restrict__ fbw,  float* __restrict__ dcw,
                           float* __restrict__ lift) {
  const double PI = 3.14159265358979323846;
  int tid = blockIdx.x * blockDim.x + threadIdx.x;
  int nth = gridDim.x * blockDim.x;

  for (int e = tid; e < NTILE * KSTEPS * 64; e += nth) {   // DFT twiddles
    int t = e / (KSTEPS * 64);
    int r = e % (KSTEPS * 64);
    int s = r >> 6, q = r & 63, lane = q >> 1, j = q & 1;
    int k = s * 4 + ((lane >= 16) ? 2 : 0) + j;
    int n = t * 16 + (lane & 15);
    double ang = 2.0 * PI * (double)((k * n) & (NFFT - 1)) / (double)NFFT;
    cosw[e] = (float)cos(ang);
    sinw[e] = (float)sin(ang);
  }
  for (int e = tid; e < 2 * FBSTEPS * 64; e += nth) {      // mel filterbank
    int t = e / (FBSTEPS * 64);
    int r = e % (FBSTEPS * 64);
    int s = r >> 6, q = r & 63, lane = q >> 1, j = q & 1;
    int k = s * 4 + ((lane >= 16) ? 2 : 0) + j;            // spectral bin
    int n = t * 16 + (lane & 15);                          // filter
    fbw[e] = (k < NBINS && n < NFILT) ? fb_val(k, n) : 0.f;
  }
  for (int e = tid; e < DCSTEPS * 64; e += nth) {          // DCT (B = dctm^T)
    int s = e >> 6, q = e & 63, lane = q >> 1, j = q & 1;
    int k = s * 4 + ((lane >= 16) ? 2 : 0) + j;            // filter
    int n = lane & 15;                                     // cepstrum
    dcw[e] = (k < NFILT && n < NUMCEP) ? dct_val(n, k) : 0.f;
  }
  for (int e = tid; e < NUMCEP; e += nth)
    lift[e] = (float)(1.0 + 11.0 * sin(PI * (double)e / 22.0));
}

__global__ void __launch_bounds__(128)
mfcc_main(const float* __restrict__ signals, const int* __restrict__ lengths,
          const float* __restrict__ cosw, const float* __restrict__ sinw,
          const float* __restrict__ fbw,  const float* __restrict__ dcw,
          const float* __restrict__ lift, float* __restrict__ out) {
  __shared__ float s_frames[16 * FROW];
  __shared__ float s_pspec [16 * PROW];
  __shared__ float s_feat  [16 * 28];
  __shared__ float s_energy[16];
  __shared__ float s_redA  [128];
  __shared__ float s_redE  [128];

  const int tid   = threadIdx.x;
  const int ftile = blockIdx.x;
  const int b     = blockIdx.y;
  const int len   = lengths[b];
  const long base = (long)ftile * 16 * FSTEP;
  const float* sig = signals + (long)b * TLEN;
  const int nfr = (len <= FLEN) ? 1 : 1 + (len - FLEN + FSTEP - 1) / FSTEP;

  // ---- Phase 1: fused pre-emphasis + length mask + framing -> LDS ----
  for (int e = tid; e < 16 * FLEN; e += 128) {
    int f = e / FLEN, k = e - f * FLEN;
    long t = base + (long)f * FSTEP + k;
    float v = 0.f;
    if (t < (long)len) {
      float cur = sig[t];
      v = (t == 0) ? cur : fmaf(-PREEMPH, sig[t - 1], cur);
    }
    s_frames[f * FROW + k] = v;
  }
  __syncthreads();

  // ---- Phase 2: 400-tap real DFT, 2 freq-tiles per wave pass ----
  // 4 independent WMMA accumulator chains hide D->C accumulation latency.
  const int lane = tid & 31;
  const int wv   = tid >> 5;
  const int arow = lane & 15;
  const int kadd = (lane >> 4) << 1;
  for (int p = wv; p < 8; p += 4) {              // wave-uniform -> EXEC all ones
    int t0 = 2 * p, t1 = 2 * p + 1;
    v8f aC0 = {0.f,0.f,0.f,0.f,0.f,0.f,0.f,0.f};
    v8f aS0 = {0.f,0.f,0.f,0.f,0.f,0.f,0.f,0.f};
    v8f aC1 = {0.f,0.f,0.f,0.f,0.f,0.f,0.f,0.f};
    v8f aS1 = {0.f,0.f,0.f,0.f,0.f,0.f,0.f,0.f};
    const v2f* cw0 = (const v2f*)cosw + (size_t)t0 * KSTEPS * 32;
    const v2f* sw0 = (const v2f*)sinw + (size_t)t0 * KSTEPS * 32;
    const v2f* cw1 = (const v2f*)cosw + (size_t)t1 * KSTEPS * 32;
    const v2f* sw1 = (const v2f*)sinw + (size_t)t1 * KSTEPS * 32;
    for (int s = 0; s < KSTEPS; ++s) {
      v2f a = *(const v2f*)&s_frames[arow * FROW + s * 4 + kadd];
      v2f b0 = cw0[s * 32 + lane];
      v2f b1 = sw0[s * 32 + lane];
      v2f b2 = cw1[s * 32 + lane];
      v2f b3 = sw1[s * 32 + lane];
      aC0 = __builtin_amdgcn_wmma_f32_16x16x4_f32(false, a, false, b0, (short)0, aC0, false, false);
      aS0 = __builtin_amdgcn_wmma_f32_16x16x4_f32(false, a, false, b1, (short)0, aS0, false, false);
      aC1 = __builtin_amdgcn_wmma_f32_16x16x4_f32(false, a, false, b2, (short)0, aC1, false, false);
      aS1 = __builtin_amdgcn_wmma_f32_16x16x4_f32(false, a, false, b3, (short)0, aS1, false, false);
    }
    int n0 = t0 * 16 + (lane & 15);
    int n1 = t1 * 16 + (lane & 15);
    int mhi = (lane >> 4) * 8;
    #pragma unroll
    for (int j = 0; j < 8; ++j) {                // bins 0..255: no guard needed
      float r0 = aC0[j], i0 = aS0[j];
      float r1 = aC1[j], i1 = aS1[j];
      s_pspec[(mhi + j) * PROW + n0] = (r0 * r0 + i0 * i0) * (1.0f / NFFT);
      s_pspec[(mhi + j) * PROW + n1] = (r1 * r1 + i1 * i1) * (1.0f / NFFT);
    }
  }
  __syncthreads();

  // ---- Phase 2b: Nyquist bin (VALU) + frame energy, 8 threads/frame ----
  {
    int f8 = tid >> 3, s8 = tid & 7;
    float alt = 0.f, esum = 0.f;
    for (int k = s8; k < FLEN; k += 8) alt  += s_frames[f8 * FROW + k];
    for (int i = s8; i < 256;  i += 8) esum += s_pspec [f8 * PROW + i];
    s_redA[tid] = (s8 & 1) ? -alt : alt;         // (-1)^k sign is per-thread const
    s_redE[tid] = esum;
  }
  __syncthreads();
  if (tid < 16) {
    float a = 0.f, e = 0.f;
    #pragma unroll
    for (int s = 0; s < 8; ++s) { a += s_redA[tid * 8 + s]; e += s_redE[tid * 8 + s]; }
    float p256 = a * a * (1.0f / NFFT);
    s_pspec[tid * PROW + 256] = p256;
    s_pspec[tid * PROW + 257] = 0.f;             // zero K-padding for fb GEMM
    s_pspec[tid * PROW + 258] = 0.f;
    s_pspec[tid * PROW + 259] = 0.f;
    s_energy[tid] = e + p256 + EPSF;
    s_feat[tid * 28 + 26] = 0.f;                 // zero K-padding for DCT GEMM
    s_feat[tid * 28 + 27] = 0.f;
  }
  __syncthreads();

  // ---- Phase 3a: mel filterbank GEMM on WMMA (16x26, K=260), waves 0..1 ----
  if (wv < 2) {
    v8f acc = {0.f,0.f,0.f,0.f,0.f,0.f,0.f,0.f};
    const v2f* bw = (const v2f*)fbw + (size_t)wv * FBSTEPS * 32;
    for (int s = 0; s < FBSTEPS; ++s) {
      v2f a = *(const v2f*)&s_pspec[arow * PROW + s * 4 + kadd];
      v2f bb = bw[s * 32 + lane];
      acc = __builtin_amdgcn_wmma_f32_16x16x4_f32(false, a, false, bb, (short)0, acc, false, false);
    }
    int n = wv * 16 + (lane & 15);
    if (n < NFILT) {
      int mhi = (lane >> 4) * 8;
      #pragma unroll
      for (int j = 0; j < 8; ++j)
        s_feat[(mhi + j) * 28 + n] = logf(acc[j] + EPSF);
    }
  }
  __syncthreads();

  // ---- Phase 3b: DCT GEMM on WMMA (16x13, K=28) + lifter + energy + mask ----
  if (wv == 0) {
    v8f acc = {0.f,0.f,0.f,0.f,0.f,0.f,0.f,0.f};
    const v2f* bw = (const v2f*)dcw;
    #pragma unroll
    for (int s = 0; s < DCSTEPS; ++s) {
      v2f a = *(const v2f*)&s_feat[arow * 28 + s * 4 + kadd];
      v2f bb = bw[s * 32 + lane];
      acc = __builtin_amdgcn_wmma_f32_16x16x4_f32(false, a, false, bb, (short)0, acc, false, false);
    }
    int n = lane & 15;                           // cepstral index
    if (n < NUMCEP) {
      float lf  = (n > 0) ? lift[n] : 0.f;
      int mhi = (lane >> 4) * 8;
      #pragma unroll
      for (int j = 0; j < 8; ++j) {
        int m = mhi + j;
        int Fg = ftile * 16 + m;
        if (Fg < NFRAMES) {
          float v = (n == 0) ? logf(s_energy[m]) : acc[j] * lf;
          if (Fg >= nfr) v = 0.f;
          out[((size_t)b * NFRAMES + Fg) * NUMCEP + n] = v;
        }
      }
    }
  }
  if (ftile == 0 && tid == 0)
    out[(size_t)BATCH * NFRAMES * NUMCEP + b] = (float)nfr;   // nframes tail
}

extern "C" void kernel_launch(void* const* d_in, const int* in_sizes, int n_in,
                              void* d_out, int out_size, void* d_ws, size_t ws_size,
                              hipStream_t stream) {
  const float* signals = (const float*)d_in[0];
  const int*   lengths = (const int*)d_in[1];
  float* out = (float*)d_out;

  float* ws   = (float*)d_ws;                        // ~854 KB used, L2-resident
  float* cosw = ws;
  float* sinw = cosw + (size_t)NTILE * KSTEPS * 64;
  float* fbw  = sinw + (size_t)NTILE * KSTEPS * 64;
  float* dcw  = fbw  + (size_t)2 * FBSTEPS * 64;
  float* lft  = dcw  + (size_t)DCSTEPS * 64;

  mfcc_setup<<<64, 256, 0, stream>>>(cosw, sinw, fbw, dcw, lft);
  mfcc_main<<<dim3(FTILES, BATCH), 128, 0, stream>>>(signals, lengths,
                                                     cosw, sinw, fbw, dcw, lft, out);
}